// CRF_14113262534666
// MI455X (gfx1250) — compile-verified
//
#include <hip/hip_runtime.h>

// Problem constants (from reference)
#define BB 64
#define TT 2048
#define FF 256
#define UU 32

// LDS row stride (floats) for the transposed weight tile:
// 260 % 64 == 4  -> writes are at worst 2-way bank conflicted,
// 260*4 = 1040 bytes, 8-byte aligned -> B fragments load as ds_load_b64.
#define KSTRIDE 260

typedef __attribute__((ext_vector_type(2))) float v2f;
typedef __attribute__((ext_vector_type(8))) float v8f;

// ---------------------------------------------------------------------------
// Kernel 1: pot[b,t,u] = x[b,t,:] @ kernel[:,u] + bias[u] (+ boundaries)
// One wave computes a 16-row x 32-col tile with V_WMMA_F32_16X16X4_F32
// (K-loop of 64 steps, two accumulators sharing the A fragment).
// The 256x32 weight matrix is staged transposed in LDS once per block.
//
// f32 WMMA A (16x4): lanes 0-15 hold rows M=0..15 with K=k0,k0+1;
// lanes 16-31 hold the same rows with K=k0+2,k0+3 (VGPR0=K even, VGPR1=odd).
// B (4x16) mirrors: lane half selects the K pair, lane&15 selects column N.
// C/D (16x16): VGPR r: lanes 0-15 -> row r, lanes 16-31 -> row r+8, N=lane&15.
// ---------------------------------------------------------------------------
__global__ __launch_bounds__(256)
void crf_potentials_wmma(const float* __restrict__ x,
                         const float* __restrict__ kern,
                         const float* __restrict__ bias,
                         const float* __restrict__ lb,
                         const float* __restrict__ rb,
                         float* __restrict__ pot)
{
    __shared__ float ktile[UU * KSTRIDE];           // kern^T, padded: 33.3 KB

    // Stage kernel (F,U) -> LDS transposed as ktile[u][k].
    // Flat index i: k = i>>5, u = i&31 -> global read fully coalesced.
    for (int i = threadIdx.x; i < FF * UU; i += 256) {
        const int k = i >> 5;
        const int u = i & 31;
        ktile[u * KSTRIDE + k] = kern[i];
    }
    __syncthreads();

    const int lane    = threadIdx.x & 31;
    const int wave    = threadIdx.x >> 5;
    const int rowTile = blockIdx.x * 8 + wave;      // 16-row tile index
    const int rowBase = rowTile << 4;               // global row = b*T + t
    const int halfSel = lane >> 4;                  // 0: K=k0,k0+1  1: K=k0+2,k0+3
    const int lmod    = lane & 15;

    const float* __restrict__ aRowPtr = x + (size_t)(rowBase + lmod) * FF;
    const float* __restrict__ b0Ptr   = ktile + lmod * KSTRIDE;
    const float* __restrict__ b1Ptr   = ktile + (16 + lmod) * KSTRIDE;

    v8f c0 = {0.f,0.f,0.f,0.f,0.f,0.f,0.f,0.f};
    v8f c1 = {0.f,0.f,0.f,0.f,0.f,0.f,0.f,0.f};

    #pragma unroll 4
    for (int k0 = 0; k0 < FF; k0 += 4) {
        const int kb = k0 + (halfSel << 1);         // even -> 8B aligned

        // A fragment: 2 consecutive K values of this lane's row (global b64)
        v2f a = *(const v2f*)(aRowPtr + kb);

        // B fragments from LDS (transposed layout -> contiguous ds_load_b64)
        v2f b0 = *(const v2f*)(b0Ptr + kb);
        v2f b1 = *(const v2f*)(b1Ptr + kb);

        c0 = __builtin_amdgcn_wmma_f32_16x16x4_f32(
                 false, a, false, b0, (short)0, c0, false, false);
        c1 = __builtin_amdgcn_wmma_f32_16x16x4_f32(
                 false, a, false, b1, (short)0, c1, false, false);
    }

    // Epilogue: fold bias + sequence-boundary additions, store pot.
    const float bias0 = bias[lmod];
    const float bias1 = bias[16 + lmod];
    const float lb0 = lb[lmod], lb1 = lb[16 + lmod];
    const float rb0 = rb[lmod], rb1 = rb[16 + lmod];

    #pragma unroll
    for (int r = 0; r < 8; ++r) {
        const int row = rowBase + r + (halfSel << 3);   // row = b*T + t
        const int t   = row & (TT - 1);                 // T is a power of two
        float add0 = bias0, add1 = bias1;
        if (t == 0)      { add0 += lb0; add1 += lb1; }
        if (t == TT - 1) { add0 += rb0; add1 += rb1; }
        float* outp = pot + (size_t)row * UU;
        outp[lmod]      = c0[r] + add0;
        outp[16 + lmod] = c1[r] + add1;
    }
}

// ---------------------------------------------------------------------------
// Kernel 2: Viterbi forward + backtrace. One wave32 per batch, lane = state u.
// Backpointers (T-1)*32 bytes live entirely in LDS (65.5 KB of the WGP's
// 320 KB), so the 2047-long dependent backtrace chain runs at LDS latency.
// ---------------------------------------------------------------------------
__global__ __launch_bounds__(32)
void crf_viterbi(const float* __restrict__ pot,
                 const float* __restrict__ chain,
                 int* __restrict__ tags)
{
    extern __shared__ unsigned char bp[];   // (TT-1) * UU bytes

    const int b = blockIdx.x;
    const int u = threadIdx.x;              // 0..31
    const float* __restrict__ potb = pot + (size_t)b * TT * UU;

    // Preload this lane's chain column: ch[k] = chain[k, u]
    float ch[UU];
    #pragma unroll
    for (int k = 0; k < UU; ++k) ch[k] = chain[k * UU + u];

    float state = potb[u];                  // t = 0

    for (int t = 1; t < TT; ++t) {
        float best = -3.402823466e38f;
        int   arg  = 0;
        #pragma unroll
        for (int k = 0; k < UU; ++k) {
            const float s    = __shfl(state, k, 32);   // broadcast state[k]
            const float cand = s + ch[k];
            if (cand > best) { best = cand; arg = k; } // first-max tie rule
        }
        bp[(t - 1) * UU + u] = (unsigned char)arg;
        state = best + potb[(size_t)t * UU + u];
    }

    // Argmax over lanes of final state (first index wins ties)
    float bestAll = -3.402823466e38f;
    int   lastTag = 0;
    #pragma unroll
    for (int k = 0; k < UU; ++k) {
        const float s = __shfl(state, k, 32);
        if (s > bestAll) { bestAll = s; lastTag = k; }
    }

    __syncthreads();   // single-wave WG: orders LDS stores vs. backtrace reads

    if (u == 0) {
        int tag = lastTag;
        int* __restrict__ tb = tags + (size_t)b * TT;
        tb[TT - 1] = tag;
        for (int t = TT - 2; t >= 0; --t) {
            tag = (int)bp[t * UU + tag];
            tb[t] = tag;
        }
    }
}

// ---------------------------------------------------------------------------
// Launch
// ---------------------------------------------------------------------------
extern "C" void kernel_launch(void* const* d_in, const int* in_sizes, int n_in,
                              void* d_out, int out_size, void* d_ws, size_t ws_size,
                              hipStream_t stream) {
    const float* x     = (const float*)d_in[0];   // (B, T, F)
    const float* kern  = (const float*)d_in[1];   // (F, U)
    const float* bias  = (const float*)d_in[2];   // (U,)
    const float* chain = (const float*)d_in[3];   // (U, U)
    const float* lb    = (const float*)d_in[4];   // (U,)
    const float* rb    = (const float*)d_in[5];   // (U,)

    int*   tags = (int*)d_out;                    // (B, T) int32
    float* pot  = (float*)d_ws;                   // (B, T, U) f32 scratch, 16 MB

    // GEMM: 131072 rows / (16 rows per wave * 8 waves per block) = 1024 blocks
    crf_potentials_wmma<<<1024, 256, 0, stream>>>(x, kern, bias, lb, rb, pot);

    // Viterbi: one wave32 per batch element; LDS holds backpointers
    const size_t bpBytes = (size_t)(TT - 1) * UU;  // 65504 bytes
    crf_viterbi<<<BB, 32, bpBytes, stream>>>(pot, chain, tags);
}